// Model_59313498358176
// MI455X (gfx1250) — compile-verified
//
#include <hip/hip_runtime.h>
#include <hip/hip_bf16.h>
#include <math.h>

// Problem constants (from the reference)
#define EXT_EPS   0.04f
#define DEPTH_MIN 0.1f
#define EDGE_MAX  0.1f
#define B_  8
#define H_  512
#define W_  512
#define N_  (H_*W_)              // 262144 pixels per view
#define FH_ ((H_-1)*(W_-1))      // 261121 lower triangles
#define F_  (2*FH_)              // 522242 faces per view

typedef __attribute__((ext_vector_type(2))) float v2f;
typedef __attribute__((ext_vector_type(8))) float v8f;

// ---------------------------------------------------------------------------
// Kernel 0: compose per-batch transform  Mhat = [E[:,:3]·K | E[:,3]],
// zero-padded to 16x4 so it can be used directly as a WMMA A-matrix.
// Mpad layout in ws: [b][16 rows][4 cols] -> 8*64 = 512 floats.
// ---------------------------------------------------------------------------
__global__ void build_mats(const float* __restrict__ kint,   // (B,3,3)
                           const float* __restrict__ kext,   // (B,3,4)
                           float* __restrict__ Mpad) {
    int idx = threadIdx.x;          // 512 threads, one element each
    int b = idx >> 6;
    int r = (idx >> 2) & 15;
    int c = idx & 3;
    float v = 0.0f;
    if (r < 3) {
        const float* E = kext + b*12 + r*4;
        if (c < 3) {
            const float* K = kint + b*9;
            v = E[0]*K[0*3+c] + E[1]*K[1*3+c] + E[2]*K[2*3+c];
        } else {
            v = E[3];
        }
    }
    Mpad[idx] = v;
}

// ---------------------------------------------------------------------------
// Kernel 1: per-pixel unprojection via V_WMMA_F32_16X16X4_F32.
// Each wave (32 lanes) handles 32 consecutive pixels with two WMMAs:
//   WMMA1 -> pixels base+0..15,  WMMA2 -> pixels base+16..31.
// A-matrix: lane L supplies row (L&15): VGPR0/1 = M[row][{0,1}] (lanes 0-15)
//           or M[row][{2,3}] (lanes 16-31).
// B-matrix: lanes 0-15 supply K0/K1 = (x*z, y*z) of their pixel column,
//           lanes 16-31 supply K2/K3 = (z, 1) of the same columns.
// D rows 0..2 (VGPR 0..2, lane = pixel column) are the transformed coords.
// ---------------------------------------------------------------------------
__global__ void vertex_kernel(const float* __restrict__ attr,   // (B,4,H,W)
                              const float* __restrict__ Mpad,
                              float* __restrict__ out) {
    const int b    = blockIdx.y;
    const int p    = blockIdx.x * blockDim.x + threadIdx.x;   // pixel in view
    const int lane = threadIdx.x & 31;
    const bool lo  = lane < 16;

    const int w = p & (W_ - 1);
    const int h = p >> 9;

    const size_t HW = (size_t)N_;
    const float* zc = attr + ((size_t)(4*b) + 3) * HW;
    const float z   = zc[p];
    const float zo  = __shfl_xor(z, 16, 32);   // z of partner lane (lane^16)

    // Extended image-plane coordinates (linspace(0.5,n-0.5,n), last += eps)
    const float x  = (float)w + 0.5f + ((w == W_-1) ? EXT_EPS : 0.0f);
    const float y  = (float)h + 0.5f + ((h == H_-1) ? EXT_EPS : 0.0f);
    const int  w2  = w + 16;                   // pixel p+16 is in the same row
    const float x2 = (float)w2 + 0.5f + ((w2 == W_-1) ? EXT_EPS : 0.0f);

    // A fragment (uniform per batch)
    const int row = lane & 15;
    const float* Mh = Mpad + (b << 6) + (row << 2);
    v2f A;
    A.x = lo ? Mh[0] : Mh[2];
    A.y = lo ? Mh[1] : Mh[3];

    // B fragments for the two pixel groups
    v2f B1, B2;
    B1.x = lo ? x  * z  : zo;       // K0: x*z   | K2: z   (of columns 0..15)
    B1.y = lo ? y  * z  : 1.0f;     // K1: y*z   | K3: 1
    B2.x = lo ? x2 * zo : z;        // columns = pixels base+16..31
    B2.y = lo ? y  * zo : 1.0f;

    v8f C = {};
    v8f D1 = __builtin_amdgcn_wmma_f32_16x16x4_f32(false, A, false, B1,
                                                   (short)0, C, false, false);
    v8f D2 = __builtin_amdgcn_wmma_f32_16x16x4_f32(false, A, false, B2,
                                                   (short)0, C, false, false);

    // Lane L<16 owns a column of WMMA1; lane L>=16 owns column L-16 of WMMA2.
    const float sx = __shfl_xor(D2[0], 16, 32);
    const float sy = __shfl_xor(D2[1], 16, 32);
    const float sz = __shfl_xor(D2[2], 16, 32);
    const float vx = lo ? D1[0] : sx;
    const float vy = lo ? D1[1] : sy;
    const float vz = lo ? D1[2] : sz;

    const size_t vo = ((size_t)b * N_ + p) * 3;
    out[vo + 0] = vx;
    out[vo + 1] = vy;
    out[vo + 2] = vz;

    // Attribute channels with boundary extension (matches the .at[] updates)
    const bool lastRow = (h == H_-1);
    const bool lastCol = (w == W_-1);
    const size_t ao = (size_t)B_ * N_ * 3 + vo;
#pragma unroll
    for (int c = 0; c < 3; ++c) {
        const float* ch = attr + ((size_t)(4*b) + c) * HW;
        float av = ch[p];
        if (lastRow && lastCol)      av += (av - ch[p - W_ - 1]) * EXT_EPS;
        else if (lastRow)            av += (av - ch[p - W_])     * EXT_EPS;
        else if (lastCol)            av += (av - ch[p - 1])      * EXT_EPS;
        out[ao + c] = av;
    }
}

// ---------------------------------------------------------------------------
// Kernel 2: per-face edge lengths + validity mask.
// Faces are generated analytically (no faces array needed):
//   cell (py,px): lower = (p, p+W, p+W+1), upper = (p, p+W+1, p+1).
// v-reads hit L2 (one view's v = 3 MB << 192 MB L2).
// ---------------------------------------------------------------------------
__global__ void face_kernel(const float* __restrict__ attr,
                            const float* __restrict__ vbuf,   // == out base
                            float* __restrict__ out) {
    const int b = blockIdx.y;
    const int f = blockIdx.x * blockDim.x + threadIdx.x;
    if (f >= F_) return;

    const int pcell = (f < FH_) ? f : f - FH_;
    const int py = pcell / (W_ - 1);
    const int px = pcell - py * (W_ - 1);
    const int pts = py * W_ + px;
    const int i0 = pts;
    int i1, i2;
    if (f < FH_) { i1 = pts + W_;     i2 = pts + W_ + 1; }
    else         { i1 = pts + W_ + 1; i2 = pts + 1;      }

    const float* zc = attr + ((size_t)(4*b) + 3) * (size_t)N_;
    const float d0 = zc[i0], d1 = zc[i1], d2 = zc[i2];
    const bool m0 = (d0 > DEPTH_MIN) && (d1 > DEPTH_MIN) && (d2 > DEPTH_MIN);

    const float* vb = vbuf + (size_t)b * N_ * 3;
    const float v0x = vb[(size_t)i0*3+0], v0y = vb[(size_t)i0*3+1], v0z = vb[(size_t)i0*3+2];
    const float v1x = vb[(size_t)i1*3+0], v1y = vb[(size_t)i1*3+1], v1z = vb[(size_t)i1*3+2];
    const float v2x = vb[(size_t)i2*3+0], v2y = vb[(size_t)i2*3+1], v2z = vb[(size_t)i2*3+2];

    // e = tri - roll(tri,-1): e0 = v0-v1, e1 = v1-v2, e2 = v2-v0
    const float e0x = v0x - v1x, e0y = v0y - v1y, e0z = v0z - v1z;
    const float e1x = v1x - v2x, e1y = v1y - v2y, e1z = v1z - v2z;
    const float e2x = v2x - v0x, e2y = v2y - v0y, e2z = v2z - v0z;
    const float l0 = sqrtf(e0x*e0x + e0y*e0y + e0z*e0z);
    const float l1 = sqrtf(e1x*e1x + e1y*e1y + e1z*e1z);
    const float l2 = sqrtf(e2x*e2x + e2y*e2y + e2z*e2z);
    const bool m1 = (l0 < EDGE_MAX) && (l1 < EDGE_MAX) && (l2 < EDGE_MAX);

    const size_t eo = (size_t)2 * B_ * N_ * 3 + ((size_t)b * F_ + f) * 3;
    out[eo + 0] = l0;
    out[eo + 1] = l1;
    out[eo + 2] = l2;

    const size_t mo = (size_t)2 * B_ * N_ * 3 + (size_t)B_ * F_ * 3
                    + (size_t)b * F_ + f;
    out[mo] = (m0 && m1) ? 1.0f : 0.0f;
}

// ---------------------------------------------------------------------------
extern "C" void kernel_launch(void* const* d_in, const int* in_sizes, int n_in,
                              void* d_out, int out_size, void* d_ws, size_t ws_size,
                              hipStream_t stream) {
    const float* attr = (const float*)d_in[0];   // (B,4,H,W) f32
    const float* kint = (const float*)d_in[1];   // (B,3,3)  f32
    const float* kext = (const float*)d_in[2];   // (B,3,4)  f32
    float* out  = (float*)d_out;
    float* Mpad = (float*)d_ws;                  // 512 floats

    build_mats<<<1, 512, 0, stream>>>(kint, kext, Mpad);
    vertex_kernel<<<dim3(N_ / 256, B_), 256, 0, stream>>>(attr, Mpad, out);
    face_kernel<<<dim3((F_ + 255) / 256, B_), 256, 0, stream>>>(attr, out, out);
}